// ThresholdedRNNCell_58042188038781
// MI455X (gfx1250) — compile-verified
//
#include <hip/hip_runtime.h>

// ThresholdedRNN on gfx1250 (MI455X), fp32 WMMA path + TDM-staged weights.
//
// Phase 1 (parallel): xin = x[B*T,I] @ Wi^T + bi  -> staged into d_out rnn region.
// Phase 2 (sequential): persistent kernel (16 blocks, grid barrier per step):
//     h_t = clamp(h_{t-1} + xin_t + h_{t-1} @ Wh^T + bh, 0, 1)
//   Each block owns one 64-wide N-tile; its Wh slice (64 rows x 4KB = 258KB)
//   is DMA'd into LDS ONCE by the Tensor Data Mover (padded/swizzled layout),
//   then all 512 timesteps feed B-fragments from LDS via ds_load_b64.
// Phase 3: copy d_out[:, T-1, :] into the final-hidden output region.

typedef __attribute__((ext_vector_type(2))) float v2f;
typedef __attribute__((ext_vector_type(8))) float v8f;
typedef __attribute__((ext_vector_type(4))) unsigned v4u;
typedef __attribute__((ext_vector_type(8))) int v8i;
typedef __attribute__((ext_vector_type(4))) int v4i;

#define B_ 128
#define T_ 512
#define I_ 512
#define H_ 1024

#define WAVES_PER_BLOCK 8      // 256 threads = 8 wave32
#define SCAN_BLOCKS 16         // one block per 64-wide N-tile of H=1024

// LDS layout for Wh slice: 64 rows x 1024 dwords, +2 pad dwords every 256
// stored dwords (TDM pad_interval=256dw, pad_amount=2dw).
// LDS dword index of linear element q (q = row*1024 + col):
//   idx = q + 2*(q>>8)   ->  row stride 1032 dwords, 8B alignment preserved.
#define WH_ROW_DW 1032
#define WH_LDS_BYTES (64 * WH_ROW_DW * 4)   // 264,192 B  (< 320KB WGP LDS)

__device__ __forceinline__ v2f ld2(const float* p) { return *(const v2f*)p; }

// ---------------------------------------------------------------------------
// Phase 1: xin[m][n] = sum_k x[m][k] * Wi[n][k] + bi[n],  m = b*T + t
// ---------------------------------------------------------------------------
__global__ void __launch_bounds__(256) rnn_proj_kernel(
    const float* __restrict__ x, const float* __restrict__ Wi,
    const float* __restrict__ bi, float* __restrict__ out) {
  const int lane = threadIdx.x & 31;
  const int wave = threadIdx.x >> 5;
  const int half = lane >> 4;     // 0: K pair {k,k+1}, 1: {k+2,k+3}
  const int l16  = lane & 15;

  const int m0 = (blockIdx.x * WAVES_PER_BLOCK + wave) * 16;
  const int n0 = blockIdx.y * 64;

  const float* arow = x  + (size_t)(m0 + l16) * I_;
  const float* bc0  = Wi + (size_t)(n0 +  0 + l16) * I_;
  const float* bc1  = Wi + (size_t)(n0 + 16 + l16) * I_;
  const float* bc2  = Wi + (size_t)(n0 + 32 + l16) * I_;
  const float* bc3  = Wi + (size_t)(n0 + 48 + l16) * I_;

  v8f acc0 = {}, acc1 = {}, acc2 = {}, acc3 = {};
#pragma unroll 2
  for (int k = 0; k < I_; k += 4) {
    const int kk = k + 2 * half;
    v2f a  = ld2(arow + kk);
    v2f f0 = ld2(bc0 + kk);
    v2f f1 = ld2(bc1 + kk);
    v2f f2 = ld2(bc2 + kk);
    v2f f3 = ld2(bc3 + kk);
    acc0 = __builtin_amdgcn_wmma_f32_16x16x4_f32(false, a, false, f0, (short)0, acc0, false, false);
    acc1 = __builtin_amdgcn_wmma_f32_16x16x4_f32(false, a, false, f1, (short)0, acc1, false, false);
    acc2 = __builtin_amdgcn_wmma_f32_16x16x4_f32(false, a, false, f2, (short)0, acc2, false, false);
    acc3 = __builtin_amdgcn_wmma_f32_16x16x4_f32(false, a, false, f3, (short)0, acc3, false, false);
  }

#pragma unroll
  for (int i = 0; i < 8; ++i) {
    const int row = m0 + i + 8 * half;
    float* orow = out + (size_t)row * H_;
    const int c = n0 + l16;
    orow[c +  0] = acc0[i] + bi[c +  0];
    orow[c + 16] = acc1[i] + bi[c + 16];
    orow[c + 32] = acc2[i] + bi[c + 32];
    orow[c + 48] = acc3[i] + bi[c + 48];
  }
}

// ---------------------------------------------------------------------------
// Phase 2 helpers
// ---------------------------------------------------------------------------
__device__ __forceinline__ void grid_barrier(unsigned* ctr, int t) {
  __syncthreads();
  __threadfence();  // make this block's global stores visible device-wide
  if (threadIdx.x == 0) {
    __hip_atomic_fetch_add(&ctr[t], 1u, __ATOMIC_ACQ_REL, __HIP_MEMORY_SCOPE_AGENT);
    while (__hip_atomic_load(&ctr[t], __ATOMIC_ACQUIRE, __HIP_MEMORY_SCOPE_AGENT)
           < (unsigned)SCAN_BLOCKS) {
      __builtin_amdgcn_s_sleep(2);
    }
  }
  __syncthreads();
}

// DMA the block's Wh slice (rows n0..n0+63, full K=1024) into LDS with the
// Tensor Data Mover, using the padded layout described above.
__device__ __forceinline__ void load_wh_to_lds(const float* Wh, int n0,
                                               float* whs) {
#if __has_builtin(__builtin_amdgcn_tensor_load_to_lds)
  if ((threadIdx.x >> 5) == 0) {
    const unsigned long long ga =
        (unsigned long long)(const void*)(Wh + (size_t)n0 * H_);
    // ---- D# group 0: count=1, lds_addr=0, global_addr, type=2 ----
    v4u g0;
    g0[0] = 1u;                                  // count=1 valid descriptor
    g0[1] = 0u;                                  // lds_addr (bytes)
    g0[2] = (unsigned)(ga & 0xFFFFFFFFull);      // global_addr[31:0]
    g0[3] = (unsigned)((ga >> 32) & 0x1FFFFFFull) | (2u << 30);  // [56:32]|type=2
    // ---- D# group 1 ----
    v8i g1;
    g1[0] = (int)((2u << 16)        // data_size = 4B
                | (1u << 20)        // pad_enable
                | (7u << 22)        // pad_interval: 256 dwords
                | (1u << 25));      // pad_amount: 2 dwords
    g1[1] = (int)((unsigned)(H_ & 0xFFFF) << 16);        // tensor_dim0.lo=1024
    g1[2] = (int)((unsigned)(H_) >> 16) | (64 << 16);    // dim0.hi | tensor_dim1=64
    g1[3] = (int)((unsigned)(H_ & 0xFFFF) << 16);        // dim1.hi=0 | tile_dim0=1024
    g1[4] = 64;                                          // tile_dim1=64, tile_dim2=0
    g1[5] = H_;                                          // tensor_dim0_stride = 1024
    g1[6] = 0;
    g1[7] = 0;
    v4i gz = {};
#if defined(__clang_major__) && (__clang_major__ >= 23)
    v8i gz8 = {};
    __builtin_amdgcn_tensor_load_to_lds(g0, g1, gz, gz, gz8, 0);
#else
    __builtin_amdgcn_tensor_load_to_lds(g0, g1, gz, gz, 0);
#endif
    __builtin_amdgcn_s_wait_tensorcnt(0);
  }
#else
  // Fallback: cooperative copy into the identical padded layout.
  for (int idx = threadIdx.x; idx < 64 * H_; idx += 256) {
    const int r = idx >> 10, q = idx & (H_ - 1);
    whs[(unsigned)(r * H_ + q) + 2u * ((unsigned)(r * H_ + q) >> 8)] =
        Wh[(size_t)(n0 + r) * H_ + q];
  }
#endif
  __syncthreads();
}

// ---------------------------------------------------------------------------
// Phase 2: persistent scan. Block b owns N-tile [b*64, b*64+64);
// wave w owns M-tile [w*16, w*16+16). B-fragments come from LDS.
// ---------------------------------------------------------------------------
__global__ void __launch_bounds__(256) rnn_scan_kernel(
    const float* __restrict__ Wh, const float* __restrict__ bh,
    const float* __restrict__ h0, float* __restrict__ out,
    unsigned* __restrict__ barrier_ctr) {
  extern __shared__ float whs[];   // WH_LDS_BYTES

  const int lane = threadIdx.x & 31;
  const int wave = threadIdx.x >> 5;
  const int half = lane >> 4;
  const int l16  = lane & 15;

  const int m0 = wave * 16;          // batch tile (8 waves cover M=128)
  const int n0 = blockIdx.x * 64;    // hidden tile (16 blocks cover H=1024)

  load_wh_to_lds(Wh, n0, whs);

  // LDS base (dwords) of each B sub-tile's row for this lane:
  // row r = s*16 + l16, element kk at dword r*1032 + kk + 2*(kk>>8).
  const float* lb0 = whs + (unsigned)(( 0 + l16) * WH_ROW_DW);
  const float* lb1 = whs + (unsigned)((16 + l16) * WH_ROW_DW);
  const float* lb2 = whs + (unsigned)((32 + l16) * WH_ROW_DW);
  const float* lb3 = whs + (unsigned)((48 + l16) * WH_ROW_DW);

  for (int t = 0; t < T_; ++t) {
    // A row = h_{t-1}[m0+l16][:]  (t==0: broadcast h0 row)
    const float* arow = (t == 0)
        ? h0
        : (out + ((size_t)(m0 + l16) * T_ + (t - 1)) * H_);

    v8f acc0 = {}, acc1 = {}, acc2 = {}, acc3 = {};
#pragma unroll 2
    for (int k = 0; k < H_; k += 4) {
      const int kk   = k + 2 * half;
      const int koff = kk + 2 * (kk >> 8);   // padded LDS offset (dwords)
      __builtin_prefetch(arow + kk + 64, 0, 1);  // global_prefetch_b8
      v2f a  = ld2(arow + kk);
      v2f f0 = *(const v2f*)(lb0 + koff);
      v2f f1 = *(const v2f*)(lb1 + koff);
      v2f f2 = *(const v2f*)(lb2 + koff);
      v2f f3 = *(const v2f*)(lb3 + koff);
      acc0 = __builtin_amdgcn_wmma_f32_16x16x4_f32(false, a, false, f0, (short)0, acc0, false, false);
      acc1 = __builtin_amdgcn_wmma_f32_16x16x4_f32(false, a, false, f1, (short)0, acc1, false, false);
      acc2 = __builtin_amdgcn_wmma_f32_16x16x4_f32(false, a, false, f2, (short)0, acc2, false, false);
      acc3 = __builtin_amdgcn_wmma_f32_16x16x4_f32(false, a, false, f3, (short)0, acc3, false, false);
    }

    // Epilogue: pre = h2h + bh + h_prev + xin ; h_new = clamp(pre, 0, 1)
#pragma unroll
    for (int i = 0; i < 8; ++i) {
      const int row = m0 + i + 8 * half;
      const size_t base = ((size_t)row * T_ + t) * H_;   // this step's slot (holds xin)
      const float acc[4] = {acc0[i], acc1[i], acc2[i], acc3[i]};
#pragma unroll
      for (int s = 0; s < 4; ++s) {
        const int c = n0 + s * 16 + l16;
        const float hp = (t == 0) ? h0[c] : out[base - H_ + c];  // h_{t-1}[row][c]
        float pre = acc[s] + bh[c] + hp + out[base + c];
        pre = fminf(fmaxf(pre, 0.0f), 1.0f);
        out[base + c] = pre;
      }
    }

    grid_barrier(barrier_ctr, t);
  }
}

// ---------------------------------------------------------------------------
// Helpers
// ---------------------------------------------------------------------------
__global__ void zero_barrier_kernel(unsigned* __restrict__ ctr, int n) {
  const int i = blockIdx.x * blockDim.x + threadIdx.x;
  if (i < n) ctr[i] = 0u;
}

__global__ void final_hidden_kernel(const float* __restrict__ rnn,
                                    float* __restrict__ hid) {
  const int i = blockIdx.x * blockDim.x + threadIdx.x;  // over B*H
  const int b = i / H_, n = i - b * H_;
  hid[i] = rnn[((size_t)b * T_ + (T_ - 1)) * H_ + n];
}

// ---------------------------------------------------------------------------
extern "C" void kernel_launch(void* const* d_in, const int* in_sizes, int n_in,
                              void* d_out, int out_size, void* d_ws, size_t ws_size,
                              hipStream_t stream) {
  const float* x  = (const float*)d_in[0];  // [B,T,I]
  const float* Wi = (const float*)d_in[1];  // [H,I]
  const float* bi = (const float*)d_in[2];  // [H]
  const float* Wh = (const float*)d_in[3];  // [H,H]
  const float* bh = (const float*)d_in[4];  // [H]
  const float* h0 = (const float*)d_in[5];  // [1,H]

  float* out = (float*)d_out;                       // [B,T,H] then [B,H]
  unsigned* barrier_ctr = (unsigned*)d_ws;          // T counters

  zero_barrier_kernel<<<(T_ + 255) / 256, 256, 0, stream>>>(barrier_ctr, T_);

  dim3 gproj((B_ * T_) / (16 * WAVES_PER_BLOCK), H_ / 64);  // 512 x 16 blocks
  rnn_proj_kernel<<<gproj, 256, 0, stream>>>(x, Wi, bi, out);

  rnn_scan_kernel<<<SCAN_BLOCKS, 256, WH_LDS_BYTES, stream>>>(
      Wh, bh, h0, out, barrier_ctr);

  final_hidden_kernel<<<(B_ * H_) / 256, 256, 0, stream>>>(
      out, out + (size_t)B_ * T_ * H_);
}